// InfoNceLoss_37563783971268
// MI455X (gfx1250) — compile-verified
//
#include <hip/hip_runtime.h>

typedef __attribute__((ext_vector_type(2))) float v2f;
typedef __attribute__((ext_vector_type(8))) float v8f;

#define NB        512          // queries / bags
#define NK        512          // keys per bag
#define ND        128          // embed dim
#define TEMP_INV  (1.0f / 0.07f)
#define BIG_NEG   (-1.0e30f)
#define NCHUNK    64           // column chunks
#define TILES_PER_WAVE 32      // 16-col tiles per wave; 8 waves * 32 * 64 chunks * 16 = 262144
#define A_STRIDE  132          // padded LDS row stride (floats) -> conflict-free b64 reads

// ---------------------------------------------------------------------------
// Kernel 1: fused f32 GEMM (v_wmma_f32_16x16x4_f32) + online logsumexp partials
// grid = (32 m-blocks, 64 chunks), block = 256 threads (8 wave32)
// B operand is double-buffered in registers at half-tile granularity so 16
// global_load_b64 are always in flight behind the WMMA stream.
// ---------------------------------------------------------------------------
__global__ __launch_bounds__(256) void infonce_gemm_lse_partial(
    const float* __restrict__ outq,     // [512,128]
    const float* __restrict__ keys,     // [512,512,128] -> flat [262144,128]
    const int*   __restrict__ selfIdx,  // [512]
    float*       __restrict__ partials) // [32][64][16][4] (gmax,gsum,bmax,bsum)
{
    __shared__ float As[16 * A_STRIDE];
    __shared__ int   sIdx[16];
    __shared__ float red[8][16][4];

    const int mblk  = blockIdx.x;
    const int chunk = blockIdx.y;
    const int b0    = mblk * 16;
    const int tid   = threadIdx.x;
    const int lane  = tid & 31;
    const int wave  = tid >> 5;

    // cooperative load of the 16x128 query tile into padded LDS
    for (int i = tid; i < 16 * ND; i += 256) {
        const int r = i >> 7;        // / 128
        const int c = i & (ND - 1);  // % 128
        As[r * A_STRIDE + c] = outq[(size_t)(b0 + r) * ND + c];
    }
    if (tid < 16) sIdx[tid] = selfIdx[b0 + tid];
    __syncthreads();

    const int h  = lane >> 4;   // half-wave select (K-pair for A/B, row+8 for C)
    const int ml = lane & 15;   // M for A, N for B/C

    // A operand is tile-invariant: pull the lane's 32 K-pairs into registers once
    const float* ap = As + ml * A_STRIDE + 2 * h;  // A: M=ml, K = 2h + {0,1} per 4-step
    v2f a[32];
#pragma unroll
    for (int kk = 0; kk < 32; ++kk) { a[kk].x = ap[4 * kk]; a[kk].y = ap[4 * kk + 1]; }

    float gmax[8], gsum[8], bmax[8], bsum[8];
#pragma unroll
    for (int j = 0; j < 8; ++j) { gmax[j] = bmax[j] = BIG_NEG; gsum[j] = bsum[j] = 0.0f; }

    const int tileBase = chunk * (8 * TILES_PER_WAVE) + wave * TILES_PER_WAVE;
    const int laneOff  = 2 * h;   // float offset into each key vector

    // lane's B pointer for tile t: key vector (16*(tileBase+t) + ml), dims laneOff + 4*kk {+1}
    const float* kp0 = keys + ((size_t)((tileBase << 4) + ml)) * ND + laneOff;

    v2f bb0[16], bb1[16];
    // prologue: first half (K dims 0..63) of tile 0
#pragma unroll
    for (int kk = 0; kk < 16; ++kk) bb0[kk] = *(const v2f*)(kp0 + 4 * kk);

    for (int t = 0; t < TILES_PER_WAVE; ++t) {
        const float* kp  = kp0 + (size_t)t * 16 * ND;                 // this tile
        const int    tn  = (t < TILES_PER_WAVE - 1) ? (t + 1) : t;    // clamped next
        const float* kpn = kp0 + (size_t)tn * 16 * ND;
        __builtin_prefetch(kpn, 0, 3);

        // issue loads for the second half (K dims 64..127) of tile t
#pragma unroll
        for (int kk = 0; kk < 16; ++kk) bb1[kk] = *(const v2f*)(kp + 64 + 4 * kk);

        // WMMA over first half while second-half loads are in flight
        v8f acc = {};
#pragma unroll
        for (int kk = 0; kk < 16; ++kk)
            acc = __builtin_amdgcn_wmma_f32_16x16x4_f32(
                false, a[kk], false, bb0[kk], (short)0, acc, false, false);

        // issue loads for the first half of tile t+1
#pragma unroll
        for (int kk = 0; kk < 16; ++kk) bb0[kk] = *(const v2f*)(kpn + 4 * kk);

        // WMMA over second half while next-tile loads are in flight
#pragma unroll
        for (int kk = 0; kk < 16; ++kk)
            acc = __builtin_amdgcn_wmma_f32_16x16x4_f32(
                false, a[kk + 16], false, bb1[kk], (short)0, acc, false, false);

        // C element j: row = j + 8h, col = ml ; fold into running logsumexps
        const int c0 = (tileBase + t) << 4;   // first column (key) of this tile
        const int nb = c0 >> 9;               // bag id of the whole tile (512 % 16 == 0)
#pragma unroll
        for (int j = 0; j < 8; ++j) {
            const int row = j + 8 * h;
            const int bg  = b0 + row;          // global query row
            float v = acc[j] * TEMP_INV;
            if ((c0 + ml) == bg * NK + sIdx[row]) v -= 1.0e12f;  // self mask

            float nm = fmaxf(gmax[j], v);
            gsum[j]  = gsum[j] * __expf(gmax[j] - nm) + __expf(v - nm);
            gmax[j]  = nm;
            if (nb == bg) {                    // element lies in this row's own bag
                float bm = fmaxf(bmax[j], v);
                bsum[j]  = bsum[j] * __expf(bmax[j] - bm) + __expf(v - bm);
                bmax[j]  = bm;
            }
        }
    }

    // merge (max,sum) pairs across the 16 lanes sharing each row set
#pragma unroll
    for (int j = 0; j < 8; ++j) {
#pragma unroll
        for (int off = 1; off < 16; off <<= 1) {
            float om = __shfl_xor(gmax[j], off);
            float os = __shfl_xor(gsum[j], off);
            float nm = fmaxf(gmax[j], om);
            gsum[j]  = gsum[j] * __expf(gmax[j] - nm) + os * __expf(om - nm);
            gmax[j]  = nm;
            om = __shfl_xor(bmax[j], off);
            os = __shfl_xor(bsum[j], off);
            nm = fmaxf(bmax[j], om);
            bsum[j]  = bsum[j] * __expf(bmax[j] - nm) + os * __expf(om - nm);
            bmax[j]  = nm;
        }
    }

    if (ml == 0) {   // lanes 0 (rows 0..7) and 16 (rows 8..15) hold the reduction
#pragma unroll
        for (int j = 0; j < 8; ++j) {
            const int row = j + 8 * h;
            red[wave][row][0] = gmax[j];
            red[wave][row][1] = gsum[j];
            red[wave][row][2] = bmax[j];
            red[wave][row][3] = bsum[j];
        }
    }
    __syncthreads();

    if (tid < 16) {  // one thread per row combines the 8 wave partials
        const int row = tid;
        float gm = BIG_NEG, gs = 0.0f, bm = BIG_NEG, bs = 0.0f;
        for (int w = 0; w < 8; ++w) {
            float om = red[w][row][0], os = red[w][row][1];
            float nm = fmaxf(gm, om);
            gs = gs * __expf(gm - nm) + os * __expf(om - nm); gm = nm;
            om = red[w][row][2]; os = red[w][row][3];
            nm = fmaxf(bm, om);
            bs = bs * __expf(bm - nm) + os * __expf(om - nm); bm = nm;
        }
        float* p = partials + (((size_t)mblk * NCHUNK + chunk) * 16 + row) * 4;
        p[0] = gm; p[1] = gs; p[2] = bm; p[3] = bs;
    }
}

// ---------------------------------------------------------------------------
// Kernel 2: combine 64 chunk partials per row -> per-row loss term
// ---------------------------------------------------------------------------
__global__ __launch_bounds__(256) void infonce_row_loss(
    const float* __restrict__ partials, float* __restrict__ losses)
{
    const int row = blockIdx.x * 256 + threadIdx.x;
    if (row >= NB) return;
    const int mblk = row >> 4, r = row & 15;

    float gm = BIG_NEG, gs = 0.0f, bm = BIG_NEG, bs = 0.0f;
    for (int chunk = 0; chunk < NCHUNK; ++chunk) {
        const float* p = partials + (((size_t)mblk * NCHUNK + chunk) * 16 + r) * 4;
        float om = p[0], os = p[1];
        float nm = fmaxf(gm, om);
        gs = gs * __expf(gm - nm) + os * __expf(om - nm); gm = nm;
        om = p[2]; os = p[3];
        nm = fmaxf(bm, om);
        bs = bs * __expf(bm - nm) + os * __expf(om - nm); bm = nm;
    }
    const float lse_total = gm + __logf(gs);
    // positives logsumexp: own-bag exps plus (B-1)*K copies of exp(0)=1
    const float CONST_ONES = (float)((NB - 1) * NK);   // 261632
    const float M2    = fmaxf(bm, 0.0f);
    const float lse_p = M2 + __logf(bs * __expf(bm - M2) + CONST_ONES * __expf(-M2));
    losses[row] = -lse_p + __logf((float)(NK - 1)) + lse_total;
}

// ---------------------------------------------------------------------------
// Kernel 3: deterministic mean of the 512 row losses
// ---------------------------------------------------------------------------
__global__ __launch_bounds__(256) void infonce_mean(
    const float* __restrict__ losses, float* __restrict__ outv)
{
    __shared__ float sm[256];
    const int tid = threadIdx.x;
    sm[tid] = losses[tid] + losses[tid + 256];
    __syncthreads();
    for (int st = 128; st > 0; st >>= 1) {
        if (tid < st) sm[tid] += sm[tid + st];
        __syncthreads();
    }
    if (tid == 0) outv[0] = sm[0] / (float)NB;
}

// ---------------------------------------------------------------------------
extern "C" void kernel_launch(void* const* d_in, const int* in_sizes, int n_in,
                              void* d_out, int out_size, void* d_ws, size_t ws_size,
                              hipStream_t stream) {
    (void)in_sizes; (void)n_in; (void)out_size; (void)ws_size;
    const float* outq    = (const float*)d_in[0];   // [512,128] f32
    const float* keys    = (const float*)d_in[1];   // [512,512,128] f32
    const int*   selfIdx = (const int*)d_in[2];     // [512] i32

    float* partials = (float*)d_ws;                            // 32*64*16*4 floats = 512 KB
    float* losses   = partials + (size_t)32 * NCHUNK * 16 * 4; // 512 floats

    dim3 grid1(32, NCHUNK);
    infonce_gemm_lse_partial<<<grid1, 256, 0, stream>>>(outq, keys, selfIdx, partials);
    infonce_row_loss<<<2, 256, 0, stream>>>(partials, losses);
    infonce_mean<<<1, 256, 0, stream>>>(losses, (float*)d_out);
}